// LearnedEquivariance_13056700580439
// MI455X (gfx1250) — compile-verified
//
#include <hip/hip_runtime.h>
#include <hip/hip_bf16.h>

// LearnedEquivariance: fuse up-to-4 circular 5-tap depthwise convs into one
// per-row 17-tap circular conv. Memory-bound: 256MB traffic @ 23.3TB/s ~ 11us.

#define D_CONST 8192
#define D_MASK  8191
#define GS      40
#define CL      17   // composite kernel length (4 x 5-tap -> 17 taps)

// Detect gfx1250 async global->LDS builtins (device pass only).
#if defined(__HIP_DEVICE_COMPILE__)
#if __has_builtin(__builtin_amdgcn_global_load_async_to_lds_b128)
#define USE_ASYNC_LDS 1
#endif
#endif

typedef int   v4i __attribute__((ext_vector_type(4)));
typedef float v4f __attribute__((ext_vector_type(4)));
typedef __attribute__((address_space(1))) v4i* as1_v4i_p;  // global
typedef __attribute__((address_space(3))) v4i* as3_v4i_p;  // LDS

// ---------------------------------------------------------------------------
// Main kernel: one block per row. Stage row in LDS (async DMA), build 17-tap
// composite kernel, apply circular conv, write coalesced NT float4 outputs.
// ---------------------------------------------------------------------------
__global__ void __launch_bounds__(256) le_conv_kernel(const float* __restrict__ x,
                                                      const int* __restrict__ g,
                                                      const float* __restrict__ kern,
                                                      const int* __restrict__ maxit_p,
                                                      float* __restrict__ out) {
  __shared__ __align__(16) float xs[D_CONST];
  __shared__ float csh[CL];

  const int row = blockIdx.x;
  const int tid = threadIdx.x;
  const float* __restrict__ xrow = x + (size_t)row * D_CONST;

  // ---- stage full row (32 KB) into LDS: 2048 x b128, 8 per thread ----
#if defined(USE_ASYNC_LDS)
  #pragma unroll
  for (int j = 0; j < 8; ++j) {
    const int idx = (tid + 256 * j) * 4;
    as1_v4i_p gsrc = (as1_v4i_p)(unsigned long long)(size_t)(xrow + idx);
    as3_v4i_p ldst = (as3_v4i_p)(unsigned)(size_t)(&xs[idx]);
    __builtin_amdgcn_global_load_async_to_lds_b128(gsrc, ldst, 0, 0);
  }
#else
  #pragma unroll
  for (int j = 0; j < 8; ++j) {
    const int idx = (tid + 256 * j) * 4;
    *(v4f*)(&xs[idx]) = *(const v4f*)(xrow + idx);
  }
#endif

  // ---- thread 0 builds the per-row composite kernel while DMA runs ----
  if (tid == 0) {
    const int gg    = g[row];
    const int sgn   = (gg > 0) - (gg < 0);
    int a           = gg < 0 ? -gg : gg;
    const int iters = a / GS;
    const int rem   = a - iters * GS;
    const int max_it = maxit_p[0];

    const float* __restrict__ kf = kern + (size_t)(rem * sgn + GS) * 5;  // final kernel
    const float* __restrict__ km = kern + (size_t)(sgn * GS + GS) * 5;   // max_k
    const float* __restrict__ ki = kern + (size_t)(GS)*5;                // id_k

    float fin[5], mk[5], idk[5];
    #pragma unroll
    for (int t = 0; t < 5; ++t) { fin[t] = kf[t]; mk[t] = km[t]; idk[t] = ki[t]; }

    float comp[CL];
    #pragma unroll
    for (int u = 0; u < CL; ++u) comp[u] = (u < 5) ? fin[u] : 0.0f;

    // three loop steps: max_k if i<iters, id_k if i<max_it, else centered delta
    for (int i = 0; i < 3; ++i) {
      const bool useMk = i < iters;
      const bool useId = i < max_it;
      float cur[5];
      #pragma unroll
      for (int t = 0; t < 5; ++t) {
        const float dlt = (t == 2) ? 1.0f : 0.0f;
        cur[t] = useMk ? mk[t] : (useId ? idk[t] : dlt);
      }
      float tmp[CL];
      #pragma unroll
      for (int u = 0; u < CL; ++u) {
        float s = 0.0f;
        #pragma unroll
        for (int t = 0; t < 5; ++t) {
          const int jj = u - t;
          if (jj >= 0) s = fmaf(cur[t], comp[jj], s);
        }
        tmp[u] = s;
      }
      #pragma unroll
      for (int u = 0; u < CL; ++u) comp[u] = tmp[u];
    }
    #pragma unroll
    for (int u = 0; u < CL; ++u) csh[u] = comp[u];
  }

#if defined(USE_ASYNC_LDS)
 #if __has_builtin(__builtin_amdgcn_s_wait_asynccnt)
  __builtin_amdgcn_s_wait_asynccnt(0);
 #else
  asm volatile("s_wait_asynccnt 0" ::: "memory");
 #endif
#endif
  __syncthreads();

  // ---- broadcast composite into registers ----
  float c[CL];
  #pragma unroll
  for (int t = 0; t < CL; ++t) c[t] = csh[t];

  // ---- circular 17-tap conv: out[d] = sum_u c[u] * x[(d+u-8) & 8191] ----
  float* __restrict__ orow = out + (size_t)row * D_CONST;
  for (int j = 0; j < 8; ++j) {
    const int s = 4 * tid + 1024 * j;  // float4 output base, 16B aligned
    float w[20];                       // window [s-8, s+12), wrap-safe chunks
    #pragma unroll
    for (int ch = 0; ch < 5; ++ch) {
      const int off = (s - 8 + 4 * ch + D_CONST) & D_MASK;  // multiple of 4
      const v4f v = *(const v4f*)(&xs[off]);
      w[4 * ch + 0] = v.x;
      w[4 * ch + 1] = v.y;
      w[4 * ch + 2] = v.z;
      w[4 * ch + 3] = v.w;
    }
    float a0 = 0.0f, a1 = 0.0f, a2 = 0.0f, a3 = 0.0f;
    #pragma unroll
    for (int t = 0; t < CL; ++t) {
      const float kt = c[t];
      a0 = fmaf(kt, w[t + 0], a0);
      a1 = fmaf(kt, w[t + 1], a1);
      a2 = fmaf(kt, w[t + 2], a2);
      a3 = fmaf(kt, w[t + 3], a3);
    }
    v4f o;
    o.x = a0; o.y = a1; o.z = a2; o.w = a3;
    // streamed output, never re-read: non-temporal store hint
    __builtin_nontemporal_store(o, (v4f*)(&orow[s]));
  }
}

// ---------------------------------------------------------------------------
// Reduction: max_it = max over batch of |g| / GROUP_SIZE (1 block, no atomics)
// ---------------------------------------------------------------------------
__global__ void __launch_bounds__(256) le_maxit_kernel(const int* __restrict__ g,
                                                       int n,
                                                       int* __restrict__ out_maxit) {
  __shared__ int red[256];
  const int tid = threadIdx.x;
  int m = 0;
  for (int i = tid; i < n; i += 256) {
    int a = g[i];
    a = a < 0 ? -a : a;
    const int it = a / GS;
    m = it > m ? it : m;
  }
  red[tid] = m;
  __syncthreads();
  for (int s = 128; s > 0; s >>= 1) {
    if (tid < s) {
      const int o = red[tid + s];
      if (o > red[tid]) red[tid] = o;
    }
    __syncthreads();
  }
  if (tid == 0) out_maxit[0] = red[0];
}

// ---------------------------------------------------------------------------
extern "C" void kernel_launch(void* const* d_in, const int* in_sizes, int n_in,
                              void* d_out, int out_size, void* d_ws, size_t ws_size,
                              hipStream_t stream) {
  const float* x    = (const float*)d_in[0];  // [B, D] fp32
  const int*   g    = (const int*)d_in[1];    // [B] int32
  const float* kern = (const float*)d_in[2];  // [81, 5] fp32
  float*       out  = (float*)d_out;          // [B, 1, D] fp32
  int*         maxit = (int*)d_ws;            // 4 bytes of scratch

  const int B = in_sizes[1];                  // 4096

  le_maxit_kernel<<<1, 256, 0, stream>>>(g, B, maxit);
  le_conv_kernel<<<B, 256, 0, stream>>>(x, g, kern, maxit, out);
}